// InfraSAGE_79577154060554
// MI455X (gfx1250) — compile-verified
//
#include <hip/hip_runtime.h>

typedef __attribute__((ext_vector_type(2))) float v2f;
typedef __attribute__((ext_vector_type(8))) float v8f;

#define N_NODES 100000
#define DFEAT   64
#define HFEAT   64

// ---------------------------------------------------------------- zero scratch
__global__ void InfraSAGE_zero(float* __restrict__ p, int n) {
    int i = blockIdx.x * blockDim.x + threadIdx.x;
    if (i < n) p[i] = 0.0f;
}

// ---------------------------------------------------------------- edge scatter
// 16 lanes per edge; each lane gathers a float4 of x[src] and atomically adds
// into agg[dst]. Lane 0 of each edge bumps the degree counter.
__global__ void InfraSAGE_scatter(const float* __restrict__ x,
                                  const long long* __restrict__ src,
                                  const long long* __restrict__ dst,
                                  float* __restrict__ agg,
                                  float* __restrict__ cnt,
                                  int E) {
    long long t = (long long)blockIdx.x * blockDim.x + threadIdx.x;
    int e = (int)(t >> 4);
    int q = (int)(t & 15);
    if (e >= E) return;
    int s = (int)src[e];
    int d = (int)dst[e];
    float4 v = *(const float4*)(x + (size_t)s * DFEAT + q * 4);
    float* a = agg + (size_t)d * DFEAT + q * 4;
    atomicAdd(a + 0, v.x);
    atomicAdd(a + 1, v.y);
    atomicAdd(a + 2, v.z);
    atomicAdd(a + 3, v.w);
    if (q == 0) atomicAdd(cnt + d, 1.0f);
}

// ---------------------------------------------------------------- fused node op
// One wave per 16-node tile: mean = agg/max(cnt,1); h = mean@Wl^T + bl + x@Wr^T
// via V_WMMA_F32_16X16X4_F32 (exact fp32); then out = sigmoid(h@Wfc^T + bfc).
__global__ __launch_bounds__(256)
void InfraSAGE_node(const float* __restrict__ x,
                    const float* __restrict__ agg,
                    const float* __restrict__ cnt,
                    const float* __restrict__ Wl,
                    const float* __restrict__ bl,
                    const float* __restrict__ Wr,
                    const float* __restrict__ Wfc,
                    const float* __restrict__ bfc,
                    float* __restrict__ out) {
    __shared__ float sWl[HFEAT * DFEAT];   // 16 KB
    __shared__ float sWr[HFEAT * DFEAT];   // 16 KB
    for (int i = threadIdx.x; i < HFEAT * DFEAT; i += 256) {
        sWl[i] = Wl[i];
        sWr[i] = Wr[i];
    }
    __syncthreads();

    const int wave = threadIdx.x >> 5;
    const int lane = threadIdx.x & 31;
    const int tile = blockIdx.x * 8 + wave;     // 16-node tile id
    const int base = tile * 16;
    if (base >= N_NODES) return;                // wave-uniform: EXEC stays all-1s

    // A-matrix 16x4 fp32 layout: lanes 0-15 hold M=lane, K=k0..k0+1;
    // lanes 16-31 hold M=lane-16, K=k0+2..k0+3.
    const int m    = lane & 15;
    const int koff = (lane < 16) ? 0 : 2;
    const int row  = base + m;

    const float inv = 1.0f / fmaxf(cnt[row], 1.0f);

    v2f aM[16], aX[16];
    const float* arow = agg + (size_t)row * DFEAT;
    const float* xrow = x   + (size_t)row * DFEAT;
#pragma unroll
    for (int ks = 0; ks < 16; ++ks) {
        float2 t0 = *(const float2*)(arow + 4 * ks + koff);
        float2 t1 = *(const float2*)(xrow + 4 * ks + koff);
        aM[ks][0] = t0.x * inv;  aM[ks][1] = t0.y * inv;
        aX[ks][0] = t1.x;        aX[ks][1] = t1.y;
    }

    float fcacc[8];
#pragma unroll
    for (int r = 0; r < 8; ++r) fcacc[r] = 0.0f;

#pragma unroll
    for (int nt = 0; nt < 4; ++nt) {
        const int n = nt * 16 + (lane & 15);    // output column owned by lane
        const float bias = bl[n];
        v8f acc;
#pragma unroll
        for (int r = 0; r < 8; ++r) acc[r] = bias;   // h starts at +b_l

        // B fragment (4x16): lane supplies W[n][k0+koff .. +1]
        const float* wlrow = sWl + n * DFEAT + koff;
#pragma unroll
        for (int ks = 0; ks < 16; ++ks) {
            v2f b; b[0] = wlrow[4 * ks]; b[1] = wlrow[4 * ks + 1];
            acc = __builtin_amdgcn_wmma_f32_16x16x4_f32(
                false, aM[ks], false, b, (short)0, acc, false, false);
        }
        const float* wrrow = sWr + n * DFEAT + koff;
#pragma unroll
        for (int ks = 0; ks < 16; ++ks) {
            v2f b; b[0] = wrrow[4 * ks]; b[1] = wrrow[4 * ks + 1];
            acc = __builtin_amdgcn_wmma_f32_16x16x4_f32(
                false, aX[ks], false, b, (short)0, acc, false, false);
        }

        // fold this N-tile into the fc dot product: fc[M] += h[M][n] * Wfc[n]
        const float wf = Wfc[n];
#pragma unroll
        for (int r = 0; r < 8; ++r) fcacc[r] += acc[r] * wf;
    }

    // Reduce over the 16 lanes of each half (the N dimension). XOR masks 1..8
    // never cross the lane-16 boundary, so the two M-halves reduce separately.
#pragma unroll
    for (int mask = 1; mask < 16; mask <<= 1) {
#pragma unroll
        for (int r = 0; r < 8; ++r)
            fcacc[r] += __shfl_xor(fcacc[r], mask, 32);
    }

    if ((lane & 15) == 0) {
        const float bb   = bfc[0];
        const int   mofs = (lane < 16) ? 0 : 8;   // C/D layout: vgpr r -> M=r (+8)
#pragma unroll
        for (int r = 0; r < 8; ++r) {
            float z = fcacc[r] + bb;
            out[base + mofs + r] = 1.0f / (1.0f + __expf(-z));
        }
    }
}

// ---------------------------------------------------------------- launcher
extern "C" void kernel_launch(void* const* d_in, const int* in_sizes, int n_in,
                              void* d_out, int out_size, void* d_ws, size_t ws_size,
                              hipStream_t stream) {
    const float*     x   = (const float*)d_in[0];
    const long long* ei  = (const long long*)d_in[1];
    const float*     Wl  = (const float*)d_in[2];
    const float*     bl  = (const float*)d_in[3];
    const float*     Wr  = (const float*)d_in[4];
    const float*     Wfc = (const float*)d_in[5];
    const float*     bfc = (const float*)d_in[6];
    float*           out = (float*)d_out;

    const int E = in_sizes[1] / 2;

    float* agg = (float*)d_ws;                       // N*D
    float* cnt = agg + (size_t)N_NODES * DFEAT;      // N

    // 1) zero scratch (ws is poisoned, and graph replays must be deterministic)
    int zn = N_NODES * DFEAT + N_NODES;
    InfraSAGE_zero<<<(zn + 255) / 256, 256, 0, stream>>>(agg, zn);

    // 2) edge scatter: 16 lanes per edge
    long long tthreads = (long long)E * 16;
    int sblocks = (int)((tthreads + 255) / 256);
    InfraSAGE_scatter<<<sblocks, 256, 0, stream>>>(x, ei, ei + E, agg, cnt, E);

    // 3) fused mean + dual WMMA GEMM + fc + sigmoid; 8 tiles (128 nodes)/block
    int ntiles  = N_NODES / 16;          // 6250 exact tiles
    int nblocks = (ntiles + 7) / 8;      // 782
    InfraSAGE_node<<<nblocks, 256, 0, stream>>>(x, agg, cnt, Wl, bl, Wr, Wfc, bfc, out);
}